// GConvLSTMBaseline_43903155699857
// MI455X (gfx1250) — compile-verified
//
#include <hip/hip_runtime.h>
#include <math.h>

typedef __attribute__((ext_vector_type(2))) float v2f;
typedef __attribute__((ext_vector_type(8))) float v8f;

#define NF 64
#define CHEB_K 7

__device__ __forceinline__ float sigm(float x) { return 1.0f / (1.0f + expf(-x)); }

// ---------------- graph preprocessing ----------------

__global__ void deg_kernel(const int* __restrict__ row, const int* __restrict__ col,
                           const float* __restrict__ ew, float* __restrict__ deg, int E) {
    int e = blockIdx.x * blockDim.x + threadIdx.x;
    if (e >= E) return;
    int r = row[e], c = col[e];
    if (r != c)
        __hip_atomic_fetch_add(&deg[r], ew[e], __ATOMIC_RELAXED, __HIP_MEMORY_SCOPE_AGENT);
}

__global__ void norm_kernel(const int* __restrict__ row, const int* __restrict__ col,
                            const float* __restrict__ ew, const float* __restrict__ deg,
                            float* __restrict__ nrm, int E) {
    int e = blockIdx.x * blockDim.x + threadIdx.x;
    if (e >= E) return;
    int r = row[e], c = col[e];
    float w  = (r == c) ? 0.0f : ew[e];
    float dr = deg[r] > 0.0f ? rsqrtf(deg[r]) : 0.0f;
    float dc = deg[c] > 0.0f ? rsqrtf(deg[c]) : 0.0f;
    nrm[e] = -dr * w * dc;
}

// out[row] += nrm[e] * z[col]  -- 16 threads per edge, float4 chunks
__global__ void prop_kernel(const int* __restrict__ row, const int* __restrict__ col,
                            const float* __restrict__ nrm, const float* __restrict__ z,
                            float* __restrict__ out, int E) {
    long gid = (long)blockIdx.x * blockDim.x + threadIdx.x;
    if (gid >= (long)E * 16) return;
    int e  = (int)(gid >> 4);
    int ch = (int)(gid & 15);
    float nm = nrm[e];
    if (nm == 0.0f) return;
    int r = row[e], c = col[e];
    const float4 z4 = *reinterpret_cast<const float4*>(z + (size_t)c * NF + ch * 4);
    float* o = out + (size_t)r * NF + ch * 4;
    __hip_atomic_fetch_add(o + 0, nm * z4.x, __ATOMIC_RELAXED, __HIP_MEMORY_SCOPE_AGENT);
    __hip_atomic_fetch_add(o + 1, nm * z4.y, __ATOMIC_RELAXED, __HIP_MEMORY_SCOPE_AGENT);
    __hip_atomic_fetch_add(o + 2, nm * z4.z, __ATOMIC_RELAXED, __HIP_MEMORY_SCOPE_AGENT);
    __hip_atomic_fetch_add(o + 3, nm * z4.w, __ATOMIC_RELAXED, __HIP_MEMORY_SCOPE_AGENT);
}

// P = 2*P - Tx0 (in place), Chebyshev recursion step
__global__ void cheb_combine(float* __restrict__ P, const float* __restrict__ Tx0, long n) {
    long gid = (long)blockIdx.x * blockDim.x + threadIdx.x;
    if (gid < n) P[gid] = 2.0f * P[gid] - Tx0[gid];
}

// dst[mat][n][k] = src[mat][k][n]  (64x64 matrices)
__global__ void transpose64(const float* __restrict__ src, float* __restrict__ dst, int nmat) {
    int gid = blockIdx.x * blockDim.x + threadIdx.x;
    if (gid >= nmat * 4096) return;
    int mat = gid >> 12, idx = gid & 4095;
    int k = idx >> 6, n = idx & 63;
    dst[(size_t)mat * 4096 + n * 64 + k] = src[gid];
}

// ---------------- WMMA fp32 GEMM: C[N x 64] (+)= A[N x 64] @ W[64 x 64] ----------------
// Bt is W transposed ([n][k]) so the B fragment is one 8-byte load.
// One wave per 16-row tile; 4 column tiles of 16; K=64 as 16 steps of 4.

__device__ __forceinline__ v8f zero8() {
    v8f z = {0.f, 0.f, 0.f, 0.f, 0.f, 0.f, 0.f, 0.f};
    return z;
}

__global__ __launch_bounds__(256)
void gemm64(const float* __restrict__ A, const float* __restrict__ Bt,
            float* __restrict__ Cm, const float* __restrict__ bias,
            int nRows, int accumulate, int addBias) {
    int wave = threadIdx.x >> 5;
    int rowBase = (blockIdx.x * 8 + wave) * 16;
    if (rowBase >= nRows) return;
    int lane = threadIdx.x & 31;
    int m = lane & 15, kh = lane >> 4;

    v2f a[16];
    const float* ap = A + (size_t)(rowBase + m) * NF + 2 * kh;
#pragma unroll
    for (int kk = 0; kk < 16; kk++) a[kk] = *reinterpret_cast<const v2f*>(ap + kk * 4);

    for (int nt = 0; nt < 4; nt++) {
        int n = nt * 16 + m;
        v8f acc;
        if (accumulate) {
#pragma unroll
            for (int v = 0; v < 8; v++) acc[v] = Cm[(size_t)(rowBase + v + 8 * kh) * NF + n];
        } else {
            acc = zero8();
        }
        const float* bp = Bt + (size_t)n * NF + 2 * kh;
#pragma unroll
        for (int kk = 0; kk < 16; kk++) {
            v2f b = *reinterpret_cast<const v2f*>(bp + kk * 4);
            acc = __builtin_amdgcn_wmma_f32_16x16x4_f32(false, a[kk], false, b,
                                                        (short)0, acc, false, false);
        }
        float bv = addBias ? bias[n] : 0.0f;
#pragma unroll
        for (int v = 0; v < 8; v++)
            Cm[(size_t)(rowBase + v + 8 * kh) * NF + n] = acc[v] + bv;
    }
}

// ---------------- fused LSTM gates (8 WMMA GEMM contributions + elementwise) ----------------
// Wt8 holds transposed [wx_i, wh_i, wx_f, wh_f, wx_c, wh_c, wx_o, wh_o]

__device__ __forceinline__ v8f mm2(const v2f* ax, const v2f* ah,
                                   const float* BtX, const float* BtH, int n, int kh) {
    v8f acc = zero8();
    const float* bp = BtX + (size_t)n * NF + 2 * kh;
#pragma unroll
    for (int kk = 0; kk < 16; kk++) {
        v2f b = *reinterpret_cast<const v2f*>(bp + kk * 4);
        acc = __builtin_amdgcn_wmma_f32_16x16x4_f32(false, ax[kk], false, b,
                                                    (short)0, acc, false, false);
    }
    bp = BtH + (size_t)n * NF + 2 * kh;
#pragma unroll
    for (int kk = 0; kk < 16; kk++) {
        v2f b = *reinterpret_cast<const v2f*>(bp + kk * 4);
        acc = __builtin_amdgcn_wmma_f32_16x16x4_f32(false, ah[kk], false, b,
                                                    (short)0, acc, false, false);
    }
    return acc;
}

__global__ __launch_bounds__(256)
void gates64(const float* __restrict__ X, const float* __restrict__ Hs,
             const float* __restrict__ Cs, const float* __restrict__ Wt8,
             const float* __restrict__ wci, const float* __restrict__ wcf,
             const float* __restrict__ wco,
             const float* __restrict__ bi, const float* __restrict__ bf_,
             const float* __restrict__ bc, const float* __restrict__ bo,
             float* __restrict__ Hout, float* __restrict__ Cout, int nRows) {
    int wave = threadIdx.x >> 5;
    int rowBase = (blockIdx.x * 8 + wave) * 16;
    if (rowBase >= nRows) return;
    int lane = threadIdx.x & 31;
    int m = lane & 15, kh = lane >> 4;

    v2f ax[16], ah[16];
    const float* xp = X  + (size_t)(rowBase + m) * NF + 2 * kh;
    const float* hp = Hs + (size_t)(rowBase + m) * NF + 2 * kh;
#pragma unroll
    for (int kk = 0; kk < 16; kk++) {
        ax[kk] = *reinterpret_cast<const v2f*>(xp + kk * 4);
        ah[kk] = *reinterpret_cast<const v2f*>(hp + kk * 4);
    }

    for (int nt = 0; nt < 4; nt++) {
        int n = nt * 16 + m;
        v8f gi = mm2(ax, ah, Wt8 + 0 * 4096, Wt8 + 1 * 4096, n, kh);
        v8f gf = mm2(ax, ah, Wt8 + 2 * 4096, Wt8 + 3 * 4096, n, kh);
        v8f gc = mm2(ax, ah, Wt8 + 4 * 4096, Wt8 + 5 * 4096, n, kh);
        v8f go = mm2(ax, ah, Wt8 + 6 * 4096, Wt8 + 7 * 4096, n, kh);

        float wciN = wci[n], wcfN = wcf[n], wcoN = wco[n];
        float biN = bi[n], bfN = bf_[n], bcN = bc[n], boN = bo[n];
#pragma unroll
        for (int v = 0; v < 8; v++) {
            size_t off = (size_t)(rowBase + v + 8 * kh) * NF + n;
            float cv = Cs[off];
            float iv = sigm(gi[v] + wciN * cv + biN);
            float fv = sigm(gf[v] + wcfN * cv + bfN);
            float tv = tanhf(gc[v] + bcN);
            float cn = fv * cv + iv * tv;
            float ov = sigm(go[v] + wcoN * cn + boN);
            Hout[off] = ov * tanhf(cn);
            Cout[off] = cn;
        }
    }
}

// ---------------- host orchestration ----------------

extern "C" void kernel_launch(void* const* d_in, const int* in_sizes, int n_in,
                              void* d_out, int out_size, void* d_ws, size_t ws_size,
                              hipStream_t stream) {
    const float* x     = (const float*)d_in[0];
    const int*   ei    = (const int*)d_in[1];
    const float* ew    = (const float*)d_in[2];
    const float* Hs    = (const float*)d_in[3];
    const float* Cs    = (const float*)d_in[4];
    const float* chebW = (const float*)d_in[5];
    const float* chebB = (const float*)d_in[6];
    const float* wx[4] = {(const float*)d_in[7],  (const float*)d_in[8],
                          (const float*)d_in[9],  (const float*)d_in[10]};  // i,f,c,o
    const float* wh[4] = {(const float*)d_in[11], (const float*)d_in[12],
                          (const float*)d_in[13], (const float*)d_in[14]};  // i,f,c,o
    const float* wci = (const float*)d_in[15];
    const float* wcf = (const float*)d_in[16];
    const float* wco = (const float*)d_in[17];
    const float* bi  = (const float*)d_in[18];
    const float* bf_ = (const float*)d_in[19];
    const float* bc  = (const float*)d_in[20];
    const float* bo  = (const float*)d_in[21];

    int Nn = in_sizes[0] / NF;
    int E  = in_sizes[1] / 2;
    const int* row = ei;
    const int* col = ei + E;
    size_t NFs = (size_t)Nn * NF;

    float* ws  = (float*)d_ws;
    float* deg = ws;
    float* nrm = deg + Nn;
    float* b0  = nrm + E;
    float* b1  = b0 + NFs;
    float* b2  = b1 + NFs;
    float* Xb  = b2 + NFs;
    float* Wt  = Xb + NFs;   // 15 transposed 64x64 matrices

    const int T = 256;
    int eBlocks = (E + T - 1) / T;
    long pThreads = (long)E * 16;
    int pBlocks = (int)((pThreads + T - 1) / T);
    long nElems = (long)NFs;
    int cBlocks = (int)((nElems + T - 1) / T);
    int gBlocks = ((Nn + 15) / 16 + 7) / 8;

    // --- degrees + edge norms ---
    hipMemsetAsync(deg, 0, (size_t)Nn * sizeof(float), stream);
    deg_kernel<<<eBlocks, T, 0, stream>>>(row, col, ew, deg, E);
    norm_kernel<<<eBlocks, T, 0, stream>>>(row, col, ew, deg, nrm, E);

    // --- transpose weights: slots 0..6 cheb, 7..14 = xi,hi,xf,hf,xc,hc,xo,ho ---
    transpose64<<<(7 * 4096 + T - 1) / T, T, 0, stream>>>(chebW, Wt, 7);
    for (int g = 0; g < 4; g++) {
        transpose64<<<16, T, 0, stream>>>(wx[g], Wt + (size_t)(7 + 2 * g) * 4096, 1);
        transpose64<<<16, T, 0, stream>>>(wh[g], Wt + (size_t)(8 + 2 * g) * 4096, 1);
    }

    // --- Chebyshev: X = sum_k T_k(Lhat) x @ W[k] + b ---
    gemm64<<<gBlocks, T, 0, stream>>>(x, Wt, Xb, chebB, Nn, 0, 0);           // k=0

    hipMemsetAsync(b1, 0, NFs * sizeof(float), stream);
    prop_kernel<<<pBlocks, T, 0, stream>>>(row, col, nrm, x, b1, E);          // Tx1
    gemm64<<<gBlocks, T, 0, stream>>>(b1, Wt + 4096, Xb, chebB, Nn, 1, 0);   // k=1

    float* bufs[3] = {b0, b1, b2};
    const float* Tx0 = x;
    float* Tx1 = b1;
    for (int k = 2; k < CHEB_K; k++) {
        float* P = bufs[k % 3];
        hipMemsetAsync(P, 0, NFs * sizeof(float), stream);
        prop_kernel<<<pBlocks, T, 0, stream>>>(row, col, nrm, Tx1, P, E);
        cheb_combine<<<cBlocks, T, 0, stream>>>(P, Tx0, nElems);              // P = 2P - Tx0
        gemm64<<<gBlocks, T, 0, stream>>>(P, Wt + (size_t)k * 4096, Xb, chebB,
                                          Nn, 1, (k == CHEB_K - 1) ? 1 : 0);
        Tx0 = Tx1;
        Tx1 = P;
    }

    // --- fused LSTM gates: writes H_new then C_new into d_out ---
    float* Hout = (float*)d_out;
    float* Cout = Hout + NFs;
    gates64<<<gBlocks, T, 0, stream>>>(Xb, Hs, Cs, Wt + (size_t)7 * 4096,
                                       wci, wcf, wco, bi, bf_, bc, bo,
                                       Hout, Cout, Nn);
}